// InterOrganGNN_32976758899215
// MI455X (gfx1250) — compile-verified
//
#include <hip/hip_runtime.h>

// InterOrganGNN fused GAT x2 + LayerNorm for gfx1250 (MI455X).
// B*T=128 graphs, N=256 nodes, F=128 features, H=4 heads, D=32.
// All matmuls via V_WMMA_F32_16X16X4_F32 (full fp32 precision, wave32).
// k-axis kept contiguous for global-memory WMMA operands -> b64 loads.

typedef float v2f __attribute__((ext_vector_type(2)));
typedef float v8f __attribute__((ext_vector_type(8)));

#define BT 128
#define NN 256
#define FF 128
#define HH 4
#define DD 32
#define NEGV (-1e9f)

__device__ __forceinline__ float wave_max32(float v) {
#pragma unroll
  for (int o = 16; o > 0; o >>= 1) v = fmaxf(v, __shfl_xor(v, o, 32));
  return v;
}
__device__ __forceinline__ float wave_sum32(float v) {
#pragma unroll
  for (int o = 16; o > 0; o >>= 1) v += __shfl_xor(v, o, 32);
  return v;
}

// ---------------------------------------------------------------------------
// Wt[n][k] = W[k][n]  (128x128, 64 KB; stays L2-resident for the whole GEMM)
// ---------------------------------------------------------------------------
__global__ __launch_bounds__(256) void transpose_w(const float* __restrict__ W,
                                                   float* __restrict__ Wt) {
  int idx = blockIdx.x * 256 + threadIdx.x;
  int n = idx >> 7, k = idx & 127;
  Wt[idx] = W[k * FF + n];
}

// ---------------------------------------------------------------------------
// WhT = (X @ W)^T in head-transposed layout WhT[((bt*H+h)*D + d)*N + node].
// Computed as Wh^T = W^T @ X^T:  A = Wt (k-contiguous b64 global loads),
// B = X^T from LDS (k-contiguous, merges into ds_load_2addr_b64).
// grid = BT*16 blocks (16-node slice each), 256 threads (8 waves, one
// feature 16-tile per wave), K=128 -> 32 WMMA per wave.
// ---------------------------------------------------------------------------
__global__ __launch_bounds__(256) void gemm_xw(const float* __restrict__ X,
                                               const float* __restrict__ Wt,
                                               float* __restrict__ WhT) {
  const int tid  = threadIdx.x;
  const int lane = tid & 31;
  const int wave = tid >> 5;
  const int bt   = blockIdx.x >> 4;
  const int m0   = (blockIdx.x & 15) * 16;  // node slice

  __shared__ float As[16][FF];  // X rows m0..m0+15 (8 KB)
  const float* Xb = X + (size_t)(bt * NN + m0) * FF;
#pragma unroll
  for (int i = 0; i < 8; ++i) {
    int idx = tid + i * 256;
    As[idx >> 7][idx & 127] = Xb[idx];
  }
  __syncthreads();

  const int n0 = wave * 16;   // feature tile
  const int kh = lane >> 4;   // k-half select (K=0..1 vs K=2..3)
  const int nl = lane & 15;   // A row (feature) / B col (node) within tile
  v8f acc = {};
  for (int kk = 0; kk < FF; kk += 4) {
    v2f a = *(const v2f*)(Wt + (size_t)(n0 + nl) * FF + kk + 2 * kh);
    v2f b = *(const v2f*)&As[nl][kk + 2 * kh];
    acc = __builtin_amdgcn_wmma_f32_16x16x4_f32(false, a, false, b,
                                                (short)0, acc, false, false);
  }
  // D tile: VGPR v -> feature row n0+v (+8 for lanes 16-31), lane -> node col.
#pragma unroll
  for (int v = 0; v < 8; ++v) {
    int c = n0 + v + 8 * kh;       // global feature index
    int h = c >> 5, d = c & 31;
    WhT[((size_t)(bt * HH + h) * DD + d) * NN + m0 + nl] = acc[v];  // coalesced
  }
}

// ---------------------------------------------------------------------------
// s1/s2 per (bt, h, n): 32-length dots of Wh head slice with a[:D], a[D:].
// WhT layout -> consecutive threads read consecutive nodes (coalesced).
// ---------------------------------------------------------------------------
__global__ __launch_bounds__(256) void gat_scores(const float* __restrict__ WhT,
                                                  const float* __restrict__ a,
                                                  float* __restrict__ s1,
                                                  float* __restrict__ s2) {
  int idx = blockIdx.x * 256 + threadIdx.x;  // (bt*H + h)*N + n
  int n  = idx & (NN - 1);
  int bh = idx >> 8;                         // bt*H + h
  const float* w = WhT + (size_t)bh * DD * NN + n;
  float v1 = 0.f, v2 = 0.f;
#pragma unroll
  for (int d = 0; d < DD; ++d) {
    float x = w[(size_t)d * NN];
    v1 += x * a[d];
    v2 += x * a[DD + d];
  }
  s1[idx] = v1;
  s2[idx] = v2;
}

// ---------------------------------------------------------------------------
// Attention per (bt, h, 32-row slice): masked softmax (att lives only in LDS),
// hp = att @ Wh_head via WMMA (B from WhT, k-contiguous b64 loads),
// fused residual + ELU written in place. grid = BT*H*8 blocks of 128 threads.
// ---------------------------------------------------------------------------
__global__ __launch_bounds__(128) void gat_attn(const float* __restrict__ WhT,
                                                const float* __restrict__ s1,
                                                const float* __restrict__ s2,
                                                const int* __restrict__ mask,
                                                const float* __restrict__ Xin,
                                                float* __restrict__ Xout) {
  const int tid  = threadIdx.x;
  const int lane = tid & 31;
  const int wave = tid >> 5;
  const int mb   = blockIdx.x & 7;
  const int h    = (blockIdx.x >> 3) & (HH - 1);
  const int bt   = blockIdx.x >> 5;
  const int m0   = mb * 32;

  __shared__ float att[32][NN];  // 32 KB, never spilled to HBM
  __shared__ float s2l[NN];
  __shared__ float s1l[32];
  __shared__ int   ml[NN];
  __shared__ int   mil[32];

  for (int i = tid; i < NN; i += 128) {
    s2l[i] = s2[(bt * HH + h) * NN + i];
    ml[i]  = mask[bt * NN + i];
  }
  if (tid < 32) {
    s1l[tid] = s1[(bt * HH + h) * NN + m0 + tid];
    mil[tid] = mask[bt * NN + m0 + tid];
  }
  __syncthreads();

  // wave-per-row masked softmax; lane covers cols lane, lane+32, ...
  for (int rr = 0; rr < 8; ++rr) {
    int   r   = wave * 8 + rr;
    int   mi  = mil[r];
    float s1v = s1l[r];
    float vmax = NEGV;
#pragma unroll
    for (int q = 0; q < 8; ++q) {
      int j = lane + 32 * q;
      float e = NEGV;
      if (mi && ml[j]) {
        float t = s1v + s2l[j];
        e = (t > 0.f) ? t : 0.2f * t;  // leaky_relu 0.2
      }
      vmax = fmaxf(vmax, e);
    }
    vmax = wave_max32(vmax);
    float vsum = 0.f;
#pragma unroll
    for (int q = 0; q < 8; ++q) {
      int j = lane + 32 * q;
      float p = 0.f;
      if (mi && ml[j]) {
        float t = s1v + s2l[j];
        float e = (t > 0.f) ? t : 0.2f * t;
        p = __expf(e - vmax);
      }
      att[r][j] = p;
      vsum += p;
    }
    vsum = wave_sum32(vsum);
    float inv = (vsum > 0.f) ? 1.f / vsum : 0.f;  // fully-masked row -> zeros
#pragma unroll
    for (int q = 0; q < 8; ++q) att[r][lane + 32 * q] *= inv;
  }
  __syncthreads();

  // hp(32x32) = att(32x256) @ Wh_head(256x32); 4 tiles, one per wave.
  const int mt = wave >> 1;
  const int nt = wave & 1;
  const int kh = lane >> 4;
  const int nl = lane & 15;
  // B[k][d]: WhT row (bt,h,d) is contiguous in k (= source node j).
  const float* Bh = WhT + ((size_t)(bt * HH + h) * DD + nt * 16 + nl) * NN;
  v8f acc = {};
  for (int kk = 0; kk < NN; kk += 4) {
    v2f a = *(const v2f*)&att[mt * 16 + nl][kk + 2 * kh];
    v2f b = *(const v2f*)(Bh + kk + 2 * kh);
    acc = __builtin_amdgcn_wmma_f32_16x16x4_f32(false, a, false, b,
                                                (short)0, acc, false, false);
  }
  // fused residual + ELU (hp already zero for masked rows)
#pragma unroll
  for (int v = 0; v < 8; ++v) {
    int grow = m0 + mt * 16 + v + 8 * kh;
    int gcol = h * DD + nt * 16 + nl;
    size_t o = (size_t)(bt * NN + grow) * FF + gcol;
    float x = Xin[o] + acc[v];
    Xout[o] = (x > 0.f) ? x : expm1f(x);
  }
}

// ---------------------------------------------------------------------------
// LayerNorm over F=128 + gamma/beta + mask. Wave per row, 4 cols per lane.
// ---------------------------------------------------------------------------
__global__ __launch_bounds__(256) void layernorm_mask(const float* __restrict__ X,
                                                      const int* __restrict__ mask,
                                                      const float* __restrict__ gamma,
                                                      const float* __restrict__ beta,
                                                      float* __restrict__ Out) {
  const int lane = threadIdx.x & 31;
  const int wave = threadIdx.x >> 5;
  const int row  = blockIdx.x * 8 + wave;  // bt*N + n
  const float* x = X + (size_t)row * FF;
  float v[4];
  float s = 0.f;
#pragma unroll
  for (int q = 0; q < 4; ++q) { v[q] = x[lane + 32 * q]; s += v[q]; }
  float mu = wave_sum32(s) * (1.f / FF);
  float var = 0.f;
#pragma unroll
  for (int q = 0; q < 4; ++q) { float d = v[q] - mu; var += d * d; }
  var = wave_sum32(var) * (1.f / FF);
  float rstd = rsqrtf(var + 1e-5f);
  float mk = (mask[row] > 0) ? 1.f : 0.f;
  float* o = Out + (size_t)row * FF;
#pragma unroll
  for (int q = 0; q < 4; ++q) {
    int c = lane + 32 * q;
    o[c] = ((v[q] - mu) * rstd * gamma[c] + beta[c]) * mk;
  }
}

// ---------------------------------------------------------------------------
extern "C" void kernel_launch(void* const* d_in, const int* in_sizes, int n_in,
                              void* d_out, int out_size, void* d_ws, size_t ws_size,
                              hipStream_t stream) {
  const float* X0    = (const float*)d_in[0];
  const int*   mask  = (const int*)d_in[1];
  const float* W0    = (const float*)d_in[2];
  const float* a0    = (const float*)d_in[3];
  const float* W1    = (const float*)d_in[4];
  const float* a1    = (const float*)d_in[5];
  const float* gamma = (const float*)d_in[6];
  const float* beta  = (const float*)d_in[7];
  float* out = (float*)d_out;

  float* Xb  = (float*)d_ws;                    // 16 MB activations
  float* WhT = Xb + (size_t)BT * NN * FF;       // 16 MB Wh (head-transposed)
  float* s1  = WhT + (size_t)BT * NN * FF;      // 0.5 MB
  float* s2  = s1 + (size_t)BT * HH * NN;       // 0.5 MB
  float* Wt  = s2 + (size_t)BT * HH * NN;       // 64 KB transposed weights

  // layer 0 (reads original input for GEMM and residual)
  transpose_w<<<64, 256, 0, stream>>>(W0, Wt);
  gemm_xw<<<BT * 16, 256, 0, stream>>>(X0, Wt, WhT);
  gat_scores<<<(BT * HH * NN) / 256, 256, 0, stream>>>(WhT, a0, s1, s2);
  gat_attn<<<BT * HH * 8, 128, 0, stream>>>(WhT, s1, s2, mask, X0, Xb);
  // layer 1 (in-place: each block writes exactly the tile it read)
  transpose_w<<<64, 256, 0, stream>>>(W1, Wt);
  gemm_xw<<<BT * 16, 256, 0, stream>>>(Xb, Wt, WhT);
  gat_scores<<<(BT * HH * NN) / 256, 256, 0, stream>>>(WhT, a1, s1, s2);
  gat_attn<<<BT * HH * 8, 128, 0, stream>>>(WhT, s1, s2, mask, Xb, Xb);
  // final LayerNorm + mask
  layernorm_mask<<<(BT * NN) / 8, 256, 0, stream>>>(Xb, mask, gamma, beta, out);
}